// GCN_1597727834503
// MI455X (gfx1250) — compile-verified
//
#include <hip/hip_runtime.h>
#include <hip/hip_bf16.h>

typedef float v2f __attribute__((ext_vector_type(2)));
typedef float v8f __attribute__((ext_vector_type(8)));

#define FDIM   128
#define CDIM   32
#define BGRAPH 512

// ---------------------------------------------------------------------------
// Dense transform: out[N,128] = hin[N,128] @ W[128,128]
// Uses V_WMMA_F32_16X16X4_F32 (fp32 in / fp32 acc -> bit-compatible numerics).
// Block: 256 threads = 8 waves; each wave owns a 16-row strip and all 8
// 16-column tiles (8 x v8f accumulators). W is staged to LDS pre-swizzled
// into B-fragment order so each lane's fragment read is a contiguous b64.
// ---------------------------------------------------------------------------
__global__ __launch_bounds__(256) void gcn_gemm_wmma(
    const float* __restrict__ hin, const float* __restrict__ W,
    float* __restrict__ out, int n)
{
  // B fragments: [k4(32)][ntile(8)][lane(32)][2 floats]  = 64 KB
  __shared__ float Bl[32 * 8 * 32 * 2];
  const int t = threadIdx.x;

  // Stage W -> LDS, permuted into WMMA B-matrix lane layout:
  //   lanes 0-15 : K = {4*k4+0, 4*k4+1},  lanes 16-31 : K = {4*k4+2, 4*k4+3}
  #pragma unroll
  for (int i = 0; i < 32; ++i) {
    int q    = i * 256 + t;        // 0..8191
    int k4   = q >> 8;
    int rem  = q & 255;
    int n0t  = rem >> 5;
    int lane = rem & 31;
    int half = lane >> 4;
    int col  = n0t * 16 + (lane & 15);
    int row  = k4 * 4 + 2 * half;
    Bl[2 * q]     = W[row * FDIM + col];
    Bl[2 * q + 1] = W[(row + 1) * FDIM + col];
  }
  __syncthreads();

  const int wave = t >> 5;
  const int lane = t & 31;
  const int half = lane >> 4;
  const int m0   = blockIdx.x * 128 + wave * 16;

  // A-matrix row for this lane (clamped so WMMA runs with EXEC all-ones)
  int arow = m0 + (lane & 15);
  if (arow > n - 1) arow = n - 1;
  const float* aptr = hin + (size_t)arow * FDIM + 2 * half;

  v8f acc[8];
  #pragma unroll
  for (int j = 0; j < 8; ++j)
    #pragma unroll
    for (int v = 0; v < 8; ++v) acc[j][v] = 0.0f;

  for (int k4 = 0; k4 < 32; ++k4) {
    float2 av = *(const float2*)(aptr + k4 * 4);
    v2f a; a.x = av.x; a.y = av.y;
    #pragma unroll
    for (int nt = 0; nt < 8; ++nt) {
      float2 bv = *(const float2*)(&Bl[((k4 * 8 + nt) * 32 + lane) * 2]);
      v2f b; b.x = bv.x; b.y = bv.y;
      acc[nt] = __builtin_amdgcn_wmma_f32_16x16x4_f32(
          false, a, false, b, (short)0, acc[nt], false, false);
    }
  }

  // D layout: VGPR v -> M = v (lanes 0-15) / v+8 (lanes 16-31), N = lane&15
  #pragma unroll
  for (int nt = 0; nt < 8; ++nt) {
    int col = nt * 16 + (lane & 15);
    #pragma unroll
    for (int v = 0; v < 8; ++v) {
      int row = m0 + v + 8 * half;
      if (row < n) out[(size_t)row * FDIM + col] = acc[nt][v];
    }
  }
}

// ---------------------------------------------------------------------------
// deg[dst] += 1 over all explicit edges (self loop folded in later as +1)
// ---------------------------------------------------------------------------
__global__ void gcn_degree(const int* __restrict__ dst, float* __restrict__ deg,
                           int e_total)
{
  int i = blockIdx.x * blockDim.x + threadIdx.x;
  if (i < e_total) atomicAdd(&deg[dst[i]], 1.0f);
}

__global__ void gcn_invsqrt(float* __restrict__ deg, int n)
{
  int i = blockIdx.x * blockDim.x + threadIdx.x;
  if (i < n) deg[i] = rsqrtf(deg[i] + 1.0f);   // deg+self-loop >= 1 always
}

// ---------------------------------------------------------------------------
// One wave per edge: hout[dst] += inv[src]*inv[dst] * tmp[src]   (128 feats)
// ---------------------------------------------------------------------------
__global__ __launch_bounds__(256) void gcn_scatter(
    const float* __restrict__ tmp, const int* __restrict__ src,
    const int* __restrict__ dst, const float* __restrict__ inv,
    float* __restrict__ hout, int e_total)
{
  int gid  = blockIdx.x * 256 + threadIdx.x;
  int e    = gid >> 5;
  int lane = gid & 31;
  if (e >= e_total) return;
  int s = src[e], d = dst[e];
  float w = inv[s] * inv[d];
  float4 v = ((const float4*)(tmp + (size_t)s * FDIM))[lane];
  float* base = hout + (size_t)d * FDIM + lane * 4;
  atomicAdd(base + 0, w * v.x);
  atomicAdd(base + 1, w * v.y);
  atomicAdd(base + 2, w * v.z);
  atomicAdd(base + 3, w * v.w);
}

// ---------------------------------------------------------------------------
// h = [relu]( scatter_result + inv[i]^2 * tmp[i]  (self loop) + bias )
// ---------------------------------------------------------------------------
__global__ void gcn_finalize(float* __restrict__ hout,
                             const float* __restrict__ tmp,
                             const float* __restrict__ inv,
                             const float* __restrict__ bias,
                             int n, int relu)
{
  int idx = blockIdx.x * blockDim.x + threadIdx.x;
  if (idx >= n * FDIM) return;
  int i = idx >> 7;
  int j = idx & 127;
  float w = inv[i];
  float v = hout[idx] + w * w * tmp[idx] + bias[j];
  hout[idx] = (relu && v < 0.0f) ? 0.0f : v;
}

// ---------------------------------------------------------------------------
// Mean-pool scatter: one wave per node
// ---------------------------------------------------------------------------
__global__ __launch_bounds__(256) void gcn_pool_scatter(
    const float* __restrict__ h, const int* __restrict__ batch,
    float* __restrict__ pooled, float* __restrict__ counts, int n)
{
  int gid  = blockIdx.x * 256 + threadIdx.x;
  int i    = gid >> 5;
  int lane = gid & 31;
  if (i >= n) return;
  int g = batch[i];
  if (lane == 0) atomicAdd(&counts[g], 1.0f);
  float4 v = ((const float4*)(h + (size_t)i * FDIM))[lane];
  float* base = pooled + (size_t)g * FDIM + lane * 4;
  atomicAdd(base + 0, v.x);
  atomicAdd(base + 1, v.y);
  atomicAdd(base + 2, v.z);
  atomicAdd(base + 3, v.w);
}

// ---------------------------------------------------------------------------
// out[b,c] = (sum_j pooled[b,j] * linW[j,c]) / max(count,1) + linb[c]
// ---------------------------------------------------------------------------
__global__ void gcn_final_linear(const float* __restrict__ pooled,
                                 const float* __restrict__ counts,
                                 const float* __restrict__ linW,
                                 const float* __restrict__ linb,
                                 float* __restrict__ out)
{
  int idx = blockIdx.x * blockDim.x + threadIdx.x;
  if (idx >= BGRAPH * CDIM) return;
  int b = idx >> 5;
  int c = idx & 31;
  float cnt = counts[b];
  if (cnt < 1.0f) cnt = 1.0f;
  const float* p = pooled + (size_t)b * FDIM;
  float acc = 0.0f;
  #pragma unroll 4
  for (int j = 0; j < FDIM; ++j) acc += p[j] * linW[j * CDIM + c];
  out[idx] = acc / cnt + linb[c];
}

// ---------------------------------------------------------------------------
extern "C" void kernel_launch(void* const* d_in, const int* in_sizes, int n_in,
                              void* d_out, int out_size, void* d_ws, size_t ws_size,
                              hipStream_t stream)
{
  (void)n_in; (void)out_size; (void)ws_size;
  const float* x     = (const float*)d_in[0];
  const int*   ei    = (const int*)d_in[1];
  const int*   batch = (const int*)d_in[2];
  const float* Ws[6];
  const float* bs[6];
  for (int i = 0; i < 6; ++i) {
    Ws[i] = (const float*)d_in[3 + 2 * i];
    bs[i] = (const float*)d_in[4 + 2 * i];
  }
  const float* linW = (const float*)d_in[15];
  const float* linb = (const float*)d_in[16];
  float* out = (float*)d_out;

  const int n = in_sizes[0] / FDIM;   // 50000 nodes
  const int E = in_sizes[1] / 2;      // 600000 edges
  const int* src  = ei;
  const int* dstv = ei + E;

  // workspace layout
  char* ws = (char*)d_ws;
  size_t hbytes = (size_t)n * FDIM * sizeof(float);          // 25.6 MB
  float* T      = (float*)ws;                                 // gemm result
  float* S      = (float*)(ws + hbytes);                      // aggregated h
  float* inv    = (float*)(ws + 2 * hbytes);                  // N floats
  float* pooled = (float*)(ws + 2 * hbytes + (size_t)n * 4);  // B*128 floats
  float* counts = pooled + (size_t)BGRAPH * FDIM;             // B floats

  // edge normalization (once)
  hipMemsetAsync(inv, 0, (size_t)n * sizeof(float), stream);
  gcn_degree<<<(E + 255) / 256, 256, 0, stream>>>(dstv, inv, E);
  gcn_invsqrt<<<(n + 255) / 256, 256, 0, stream>>>(inv, n);

  const float* hin = x;
  for (int L = 0; L < 6; ++L) {
    gcn_gemm_wmma<<<(n + 127) / 128, 256, 0, stream>>>(hin, Ws[L], T, n);
    hipMemsetAsync(S, 0, hbytes, stream);
    gcn_scatter<<<(E * 32 + 255) / 256, 256, 0, stream>>>(T, src, dstv, inv, S, E);
    gcn_finalize<<<(n * FDIM + 255) / 256, 256, 0, stream>>>(
        S, T, inv, bs[L], n, (L < 5) ? 1 : 0);
    hin = S;
  }

  // global mean pool + classifier
  hipMemsetAsync(pooled, 0, (size_t)(BGRAPH * FDIM + BGRAPH) * sizeof(float), stream);
  gcn_pool_scatter<<<(n * 32 + 255) / 256, 256, 0, stream>>>(S, batch, pooled, counts, n);
  gcn_final_linear<<<(BGRAPH * CDIM + 255) / 256, 256, 0, stream>>>(
      pooled, counts, linW, linb, out);
}